// ValueDetector_63316407878009
// MI455X (gfx1250) — compile-verified
//
#include <hip/hip_runtime.h>
#include <hip/hip_bf16.h>
#include <math.h>

// Problem dims
#define B_   32
#define T_   512
#define DB_  768
#define DL_  256
#define H2_  128
#define NC_  16
#define XA_  10

typedef __attribute__((ext_vector_type(2))) float v2f;
typedef __attribute__((ext_vector_type(8))) float v8f;

__device__ __forceinline__ float sigmoidf_(float x) { return 1.0f / (1.0f + __expf(-x)); }

// ---------------------------------------------------------------------------
// Async global->LDS copy (CDNA5): GLOBAL_LOAD_ASYNC_TO_LDS_B128, ASYNCcnt.
// Builtin signature (per hipcc diagnostics): (int4 __device__*, int4 __shared__*,
// imm offset, imm cpol) with GCC-style vector int4 pointees.
// ---------------------------------------------------------------------------
#if __has_builtin(__builtin_amdgcn_global_load_async_to_lds_b128) && \
    __has_builtin(__builtin_amdgcn_s_wait_asynccnt)
#define ASYNC_LDS_BUILTIN 1
#else
#define ASYNC_LDS_BUILTIN 0
#endif

typedef int v4i_ __attribute__((vector_size(16)));
typedef __attribute__((address_space(1))) v4i_* gv4_p;
typedef __attribute__((address_space(3))) v4i_* lv4_p;

__device__ __forceinline__ void g2l_f4(const float* __restrict__ g, float* l)
{
#if ASYNC_LDS_BUILTIN
    __builtin_amdgcn_global_load_async_to_lds_b128(
        (gv4_p)(uintptr_t)g, (lv4_p)(uintptr_t)l, 0, 0);
#else
    unsigned int       laddr = (unsigned int)(uintptr_t)l;        // LDS byte offset
    unsigned long long gaddr = (unsigned long long)(uintptr_t)g;  // global address
    asm volatile("global_load_async_to_lds_b128 %0, %1, off"
                 :: "v"(laddr), "v"(gaddr) : "memory");
#endif
}

__device__ __forceinline__ void async_wait0()
{
#if ASYNC_LDS_BUILTIN
    __builtin_amdgcn_s_wait_asynccnt(0);
#else
    asm volatile("s_wait_asynccnt 0x0" ::: "memory");
#endif
}

// ---------------------------------------------------------------------------
// Generic WMMA f32 GEMM:  out(MxN) = concat(A1|A2)(MxK) @ W(NxK)^T + bias1 + bias2
// One wave per 16x16 tile; split K-loops (no per-iteration concat selects) and
// dual accumulators so two independent load->wmma chains overlap.
// Fragment layouts (V_WMMA_F32_16X16X4_F32):
//   A (16x4): lane l, vgpr v -> M = l&15, K = v + 2*(l>>4)
//   B (4x16): lane l, vgpr v -> N = l&15, K = v + 2*(l>>4)
//   D (16x16): lane l, vgpr r -> M = r + 8*(l>>4), N = l&15
// Requires K1 % 8 == 0 and K2 % 8 == 0.
// ---------------------------------------------------------------------------
__global__ __launch_bounds__(256)
void k_gemm_wmma(const float* __restrict__ A1, int K1,
                 const float* __restrict__ A2, int K2,
                 const float* __restrict__ W,
                 const float* __restrict__ bias1,
                 const float* __restrict__ bias2,
                 float* __restrict__ out, int M, int N)
{
    const int lane = threadIdx.x & 31;
    const int wave = blockIdx.x * (blockDim.x >> 5) + (threadIdx.x >> 5);
    const int ntN  = N >> 4;
    const int ntiles = (M >> 4) * ntN;
    if (wave >= ntiles) return;                 // wave-uniform
    const int m0  = (wave / ntN) << 4;
    const int n0  = (wave % ntN) << 4;
    const int l15 = lane & 15;
    const int lh  = (lane >> 4) << 1;           // 0 or 2
    const int K   = K1 + K2;

    float binit = 0.0f;
    if (bias1) binit += bias1[n0 + l15];
    if (bias2) binit += bias2[n0 + l15];
    v8f cA, cB;
#pragma unroll
    for (int r = 0; r < 8; ++r) { cA[r] = binit; cB[r] = 0.0f; }

    const float* wrow = W + (size_t)(n0 + l15) * K;
    {   // A1 segment
        const float* arow = A1 + (size_t)(m0 + l15) * K1;
        for (int k = 0; k < K1; k += 8) {
            v2f a0, b0, a1, b1;
            a0.x = arow[k + lh];     a0.y = arow[k + lh + 1];
            b0.x = wrow[k + lh];     b0.y = wrow[k + lh + 1];
            a1.x = arow[k + 4 + lh]; a1.y = arow[k + 4 + lh + 1];
            b1.x = wrow[k + 4 + lh]; b1.y = wrow[k + 4 + lh + 1];
            cA = __builtin_amdgcn_wmma_f32_16x16x4_f32(false, a0, false, b0, (short)0, cA, false, false);
            cB = __builtin_amdgcn_wmma_f32_16x16x4_f32(false, a1, false, b1, (short)0, cB, false, false);
        }
    }
    if (K2 > 0) {   // A2 segment (wave-uniform predicate)
        const float* arow = A2 + (size_t)(m0 + l15) * K2;
        const float* wro2 = wrow + K1;
        for (int k = 0; k < K2; k += 8) {
            v2f a0, b0, a1, b1;
            a0.x = arow[k + lh];     a0.y = arow[k + lh + 1];
            b0.x = wro2[k + lh];     b0.y = wro2[k + lh + 1];
            a1.x = arow[k + 4 + lh]; a1.y = arow[k + 4 + lh + 1];
            b1.x = wro2[k + 4 + lh]; b1.y = wro2[k + 4 + lh + 1];
            cA = __builtin_amdgcn_wmma_f32_16x16x4_f32(false, a0, false, b0, (short)0, cA, false, false);
            cB = __builtin_amdgcn_wmma_f32_16x16x4_f32(false, a1, false, b1, (short)0, cB, false, false);
        }
    }
    const int mrow = m0 + ((lane >> 4) << 3);
#pragma unroll
    for (int r = 0; r < 8; ++r)
        out[(size_t)(mrow + r) * N + n0 + l15] = cA[r] + cB[r];
}

// ---------------------------------------------------------------------------
// Bidirectional encoder LSTM. grid = 2 blocks (0=fwd, 1=rev), 1024 threads.
// Persistent on one WGP: h,c,gates in LDS; per-step G slice (32x512, 64KB)
// double-buffered via async global->LDS while the WMMA recurrence runs.
// LDS: h 16KB | c 16KB | gates 64KB | 2x G-stage 128KB = 224KB (< 320KB/WGP).
// ---------------------------------------------------------------------------
__global__ __launch_bounds__(1024)
void k_lstm_enc(const float* __restrict__ Gf, const float* __restrict__ Gr,
                const float* __restrict__ Whh_f, const float* __restrict__ Whh_r,
                float* __restrict__ hs)
{
    extern __shared__ float smem[];
    float* shh  = smem;           // 32*128
    float* scc  = smem + 4096;    // 32*128
    float* sg   = smem + 8192;    // 32*512
    float* bufG = smem + 24576;   // 2 * 32*512
    const int tid = threadIdx.x;
    const int dir = blockIdx.x;
    const float* G   = dir ? Gr : Gf;
    const float* Whh = dir ? Whh_r : Whh_f;

    for (int e = tid; e < 4096; e += 1024) { shh[e] = 0.0f; scc[e] = 0.0f; }

    // stage step 0
    {
        const int tt0 = dir ? (T_ - 1) : 0;
#pragma unroll
        for (int i = 0; i < 4; ++i) {
            const int f  = tid + (i << 10);   // float4 index 0..4095
            const int b  = f >> 7;
            const int c4 = f & 127;
            g2l_f4(G + (((size_t)(b * T_ + tt0)) << 9) + (c4 << 2), bufG + (f << 2));
        }
    }
    async_wait0();
    __syncthreads();

    const int lane = tid & 31;
    const int wave = tid >> 5;
    const int l15  = lane & 15;
    const int lh   = (lane >> 4) << 1;
    const int rbase = (lane >> 4) << 3;       // 0 or 8

    for (int t = 0; t < T_; ++t) {
        const int tt = dir ? (T_ - 1 - t) : t;
        float* cur = bufG + ((t & 1) << 14);
        float* nxt = bufG + (((t + 1) & 1) << 14);
        // prefetch next step's G slice into the idle buffer
        if (t + 1 < T_) {
            const int tn = dir ? (T_ - 2 - t) : (t + 1);
#pragma unroll
            for (int i = 0; i < 4; ++i) {
                const int f  = tid + (i << 10);
                const int b  = f >> 7;
                const int c4 = f & 127;
                g2l_f4(G + (((size_t)(b * T_ + tn)) << 9) + (c4 << 2), nxt + (f << 2));
            }
        }
        // gates = G_t + h @ Whh^T : one N-strip per wave, both M-halves share B
        {
            const int n0 = wave << 4;
            v8f c0, c1;
#pragma unroll
            for (int r = 0; r < 8; ++r) {
                c0[r] = cur[(rbase + r) * 512 + n0 + l15];
                c1[r] = cur[(16 + rbase + r) * 512 + n0 + l15];
            }
            const float* wr = Whh + (size_t)(n0 + l15) * H2_;
            const float* h0 = shh + l15 * H2_;
            const float* h1 = shh + (16 + l15) * H2_;
            for (int k = 0; k < H2_; k += 4) {
                v2f a0, a1, b;
                b.x  = wr[k + lh]; b.y  = wr[k + lh + 1];
                a0.x = h0[k + lh]; a0.y = h0[k + lh + 1];
                a1.x = h1[k + lh]; a1.y = h1[k + lh + 1];
                c0 = __builtin_amdgcn_wmma_f32_16x16x4_f32(false, a0, false, b, (short)0, c0, false, false);
                c1 = __builtin_amdgcn_wmma_f32_16x16x4_f32(false, a1, false, b, (short)0, c1, false, false);
            }
#pragma unroll
            for (int r = 0; r < 8; ++r) {
                sg[(rbase + r) * 512 + n0 + l15]      = c0[r];
                sg[(16 + rbase + r) * 512 + n0 + l15] = c1[r];
            }
        }
        __syncthreads();
        // elementwise cell update, gate order i,f,g,o
        for (int e = tid; e < 4096; e += 1024) {
            const int b = e >> 7, j = e & 127;
            const float* gr = sg + b * 512;
            const float iv = sigmoidf_(gr[j]);
            const float fv = sigmoidf_(gr[128 + j]);
            const float gv = tanhf(gr[256 + j]);
            const float ov = sigmoidf_(gr[384 + j]);
            const float cv = fv * scc[e] + iv * gv;
            const float hv = ov * tanhf(cv);
            scc[e] = cv; shh[e] = hv;
            hs[((size_t)b * T_ + tt) * 256 + dir * 128 + j] = hv;
        }
        async_wait0();       // next G slice resident
        __syncthreads();
    }
}

// ---------------------------------------------------------------------------
// qp/kp projections: qp[row,x] = hs[row,:] . fc1_w[x,0:256];  kp uses cols 256:512
// ---------------------------------------------------------------------------
__global__ __launch_bounds__(256)
void k_qk(const float* __restrict__ hs, const float* __restrict__ fc1,
          float* __restrict__ qp, float* __restrict__ kp)
{
    const int idx = blockIdx.x * blockDim.x + threadIdx.x;
    if (idx >= B_ * T_ * XA_) return;
    const int row = idx / XA_, x = idx % XA_;
    const float* h  = hs  + (size_t)row * 256;
    const float* wq = fc1 + (size_t)x * 512;
    const float* wk = wq + 256;
    float aq = 0.0f, ak = 0.0f;
    for (int d = 0; d < 256; ++d) { aq += h[d] * wq[d]; ak += h[d] * wk[d]; }
    qp[idx] = aq; kp[idx] = ak;
}

// ---------------------------------------------------------------------------
// Fused attention: per (b,i) row -> scores over j, masked softmax, ctx = w @ hs
// ---------------------------------------------------------------------------
__global__ __launch_bounds__(256)
void k_attn(const float* __restrict__ qp, const float* __restrict__ kp,
            const float* __restrict__ fc2, const int* __restrict__ mask,
            const float* __restrict__ hs, float* __restrict__ ctx)
{
    __shared__ float sw[512];
    __shared__ float red[256];
    __shared__ float sq[XA_];
    __shared__ float sv[XA_];
    const int row = blockIdx.x;           // b*T + i
    const int b   = row >> 9;
    const int tid = threadIdx.x;
    if (tid < XA_) { sq[tid] = qp[(size_t)row * XA_ + tid]; sv[tid] = fc2[tid]; }
    __syncthreads();

    float lmax = -3.4e38f;
    for (int j = tid; j < 512; j += 256) {
        const float* kpr = kp + (size_t)(b * 512 + j) * XA_;
        float s = 0.0f;
#pragma unroll
        for (int x = 0; x < XA_; ++x) s += sv[x] * tanhf(sq[x] + kpr[x]);
        if (mask[b * 512 + j] == 0) s = -3.4e38f;
        sw[j] = s;
        lmax = fmaxf(lmax, s);
    }
    red[tid] = lmax; __syncthreads();
    for (int off = 128; off > 0; off >>= 1) {
        if (tid < off) red[tid] = fmaxf(red[tid], red[tid + off]);
        __syncthreads();
    }
    const float mx = red[0];
    __syncthreads();
    float lsum = 0.0f;
    for (int j = tid; j < 512; j += 256) {
        const float e = __expf(sw[j] - mx);
        sw[j] = e; lsum += e;
    }
    red[tid] = lsum; __syncthreads();
    for (int off = 128; off > 0; off >>= 1) {
        if (tid < off) red[tid] += red[tid + off];
        __syncthreads();
    }
    const float inv = 1.0f / red[0];
    __syncthreads();
    // each thread owns one of the 256 hidden dims
    const float* hb = hs + (size_t)(b * 512) * 256;
    float acc = 0.0f;
    for (int j = 0; j < 512; ++j) acc += sw[j] * hb[(size_t)j * 256 + tid];
    ctx[(size_t)row * 256 + tid] = acc * inv;
}

// ---------------------------------------------------------------------------
// Decoder: 1 block, 1024 threads, 512 sequential steps.
// gates(32x1024) = y_enc_t@Wih[:, :256]^T + y_prev@Wih[:, 256:]^T + h@Whh^T + b
// as three K-loops; y_enc slice (32KB) double-buffered via async global->LDS.
// LDS: 2x y-stage 64KB | h 32KB | c 32KB | gates 128KB | y/logits 4KB = 266KB.
// ---------------------------------------------------------------------------
__global__ __launch_bounds__(1024)
void k_decoder(const float* __restrict__ yenc, const float* __restrict__ hs,
               const float* __restrict__ Wih, const float* __restrict__ Whh,
               const float* __restrict__ bih, const float* __restrict__ bhh,
               const float* __restrict__ linw, const float* __restrict__ linb,
               float* __restrict__ out)
{
    extern __shared__ float smem[];
    float* bufY = smem;           // 2 * 32*256
    float* shh  = smem + 16384;   // 32*256
    float* scc  = smem + 24576;   // 32*256
    float* sg   = smem + 32768;   // 32*1024
    float* sy   = smem + 65536;   // 32*16
    float* slg  = smem + 66048;   // 32*16
    const int tid = threadIdx.x;

    for (int e = tid; e < 8192; e += 1024) {
        const int b = e >> 8, d = e & 255;
        shh[e] = hs[((size_t)b * T_ + (T_ - 1)) * 256 + d];  // h0 = hs[:, -1, :]
        scc[e] = 0.0f;
    }
    for (int e = tid; e < 512; e += 1024) sy[e] = 0.0f;
    // stage step 0
#pragma unroll
    for (int i = 0; i < 2; ++i) {
        const int f  = tid + (i << 10);   // float4 index 0..2047
        const int b  = f >> 6;
        const int c4 = f & 63;
        g2l_f4(yenc + (((size_t)(b * T_)) << 8) + (c4 << 2), bufY + (f << 2));
    }
    async_wait0();
    __syncthreads();

    const int lane = tid & 31;
    const int wave = tid >> 5;
    const int l15  = lane & 15;
    const int lh   = (lane >> 4) << 1;
    const int rbase = (lane >> 4) << 3;

    for (int t = 0; t < T_; ++t) {
        float* cur = bufY + ((t & 1) << 13);
        float* nxt = bufY + (((t + 1) & 1) << 13);
        if (t + 1 < T_) {
#pragma unroll
            for (int i = 0; i < 2; ++i) {
                const int f  = tid + (i << 10);
                const int b  = f >> 6;
                const int c4 = f & 63;
                g2l_f4(yenc + (((size_t)(b * T_ + t + 1)) << 8) + (c4 << 2), nxt + (f << 2));
            }
        }
        // gates: 2 N-strips per wave, both M-halves share the B fragment
#pragma unroll
        for (int nn = 0; nn < 2; ++nn) {
            const int n0 = (wave + (nn << 5)) << 4;
            const float binit = bih[n0 + l15] + bhh[n0 + l15];
            v8f c0, c1;
#pragma unroll
            for (int r = 0; r < 8; ++r) { c0[r] = binit; c1[r] = binit; }
            const float* w1 = Wih + (size_t)(n0 + l15) * 272;
            const float* w2 = Whh + (size_t)(n0 + l15) * 256;
            const float* y0 = cur + l15 * 256;
            const float* y1 = cur + (16 + l15) * 256;
            for (int k = 0; k < 256; k += 4) {          // y_enc part
                v2f a0, a1, b;
                b.x  = w1[k + lh]; b.y  = w1[k + lh + 1];
                a0.x = y0[k + lh]; a0.y = y0[k + lh + 1];
                a1.x = y1[k + lh]; a1.y = y1[k + lh + 1];
                c0 = __builtin_amdgcn_wmma_f32_16x16x4_f32(false, a0, false, b, (short)0, c0, false, false);
                c1 = __builtin_amdgcn_wmma_f32_16x16x4_f32(false, a1, false, b, (short)0, c1, false, false);
            }
            const float* s0 = sy + l15 * 16;
            const float* s1 = sy + (16 + l15) * 16;
            for (int k = 0; k < 16; k += 4) {           // y_prev feedback part
                v2f a0, a1, b;
                b.x  = w1[256 + k + lh]; b.y  = w1[256 + k + lh + 1];
                a0.x = s0[k + lh];       a0.y = s0[k + lh + 1];
                a1.x = s1[k + lh];       a1.y = s1[k + lh + 1];
                c0 = __builtin_amdgcn_wmma_f32_16x16x4_f32(false, a0, false, b, (short)0, c0, false, false);
                c1 = __builtin_amdgcn_wmma_f32_16x16x4_f32(false, a1, false, b, (short)0, c1, false, false);
            }
            const float* h0 = shh + l15 * 256;
            const float* h1 = shh + (16 + l15) * 256;
            for (int k = 0; k < 256; k += 4) {          // recurrent part
                v2f a0, a1, b;
                b.x  = w2[k + lh]; b.y  = w2[k + lh + 1];
                a0.x = h0[k + lh]; a0.y = h0[k + lh + 1];
                a1.x = h1[k + lh]; a1.y = h1[k + lh + 1];
                c0 = __builtin_amdgcn_wmma_f32_16x16x4_f32(false, a0, false, b, (short)0, c0, false, false);
                c1 = __builtin_amdgcn_wmma_f32_16x16x4_f32(false, a1, false, b, (short)0, c1, false, false);
            }
#pragma unroll
            for (int r = 0; r < 8; ++r) {
                sg[(rbase + r) * 1024 + n0 + l15]      = c0[r];
                sg[(16 + rbase + r) * 1024 + n0 + l15] = c1[r];
            }
        }
        __syncthreads();
        // cell update
        for (int e = tid; e < 8192; e += 1024) {
            const int b = e >> 8, j = e & 255;
            const float* gr = sg + b * 1024;
            const float iv = sigmoidf_(gr[j]);
            const float fv = sigmoidf_(gr[256 + j]);
            const float gv = tanhf(gr[512 + j]);
            const float ov = sigmoidf_(gr[768 + j]);
            const float cv = fv * scc[e] + iv * gv;
            shh[e] = ov * tanhf(cv);
            scc[e] = cv;
        }
        __syncthreads();
        // logits = h @ lin_w^T + lin_b
        if (tid < 512) {
            const int b = tid >> 4, n = tid & 15;
            const float* hv = shh + b * 256;
            const float* wv = linw + n * 256;
            float acc = linb[n];
            for (int d = 0; d < 256; ++d) acc += hv[d] * wv[d];
            slg[tid] = acc;
        }
        __syncthreads();
        // softmax per batch row, emit output (layout (B,T,NC))
        if (tid < 32) {
            const int b = tid;
            float mx = -3.4e38f;
            for (int n = 0; n < 16; ++n) mx = fmaxf(mx, slg[b * 16 + n]);
            float ex[16]; float s = 0.0f;
            for (int n = 0; n < 16; ++n) { ex[n] = __expf(slg[b * 16 + n] - mx); s += ex[n]; }
            const float inv = 1.0f / s;
            for (int n = 0; n < 16; ++n) {
                const float p = ex[n] * inv;
                sy[b * 16 + n] = p;
                out[((size_t)b * T_ + t) * 16 + n] = p;
            }
        }
        async_wait0();       // next y_enc slice resident
        __syncthreads();
    }
}

// ---------------------------------------------------------------------------
extern "C" void kernel_launch(void* const* d_in, const int* in_sizes, int n_in,
                              void* d_out, int out_size, void* d_ws, size_t ws_size,
                              hipStream_t stream)
{
    (void)in_sizes; (void)n_in; (void)out_size; (void)ws_size;
    const float* x     = (const float*)d_in[0];
    const int*   mask  = (const int*)  d_in[1];
    const float* Wih_f = (const float*)d_in[2];
    const float* Whh_f = (const float*)d_in[3];
    const float* bih_f = (const float*)d_in[4];
    const float* bhh_f = (const float*)d_in[5];
    const float* Wih_r = (const float*)d_in[6];
    const float* Whh_r = (const float*)d_in[7];
    const float* bih_r = (const float*)d_in[8];
    const float* bhh_r = (const float*)d_in[9];
    const float* fc1_w = (const float*)d_in[10];
    const float* fc2_w = (const float*)d_in[11];
    const float* fc3_w = (const float*)d_in[12];
    const float* fc3_b = (const float*)d_in[13];
    const float* Wih_d = (const float*)d_in[14];
    const float* Whh_d = (const float*)d_in[15];
    const float* bih_d = (const float*)d_in[16];
    const float* bhh_d = (const float*)d_in[17];
    const float* lin_w = (const float*)d_in[18];
    const float* lin_b = (const float*)d_in[19];
    float* out = (float*)d_out;

    float* ws  = (float*)d_ws;
    float* Gf  = ws; ws += (size_t)16384 * 512;
    float* Gr  = ws; ws += (size_t)16384 * 512;
    float* hsb = ws; ws += (size_t)16384 * 256;
    float* qp  = ws; ws += (size_t)16384 * XA_;
    float* kp  = ws; ws += (size_t)16384 * XA_;
    float* ctx = ws; ws += (size_t)16384 * 256;
    float* ye  = ws; ws += (size_t)16384 * 256;

    // 1) hoisted input projections G = x @ Wih^T + bih + bhh (both directions)
    {
        const int M = 16384, N = 512;
        const int tiles  = (M / 16) * (N / 16);
        const int blocks = (tiles * 32 + 255) / 256;
        k_gemm_wmma<<<blocks, 256, 0, stream>>>(x, 768, nullptr, 0, Wih_f, bih_f, bhh_f, Gf, M, N);
        k_gemm_wmma<<<blocks, 256, 0, stream>>>(x, 768, nullptr, 0, Wih_r, bih_r, bhh_r, Gr, M, N);
    }
    // 2) bidirectional encoder LSTM (persistent, 2 WGPs, async double-buffered G)
    k_lstm_enc<<<2, 1024, 57344 * sizeof(float), stream>>>(Gf, Gr, Whh_f, Whh_r, hsb);
    // 3) attention projections
    {
        const int n = 16384 * XA_;
        k_qk<<<(n + 255) / 256, 256, 0, stream>>>(hsb, fc1_w, qp, kp);
    }
    // 4) fused scores + softmax + context
    k_attn<<<16384, 256, 0, stream>>>(qp, kp, fc2_w, mask, hsb, ctx);
    // 5) y_enc = [hs | ctx] @ fc3_w^T + fc3_b  (concat-aware GEMM, no copy)
    {
        const int M = 16384, N = 256;
        const int tiles  = (M / 16) * (N / 16);
        const int blocks = (tiles * 32 + 255) / 256;
        k_gemm_wmma<<<blocks, 256, 0, stream>>>(hsb, 256, ctx, 256, fc3_w, fc3_b, nullptr, ye, M, N);
    }
    // 6) decoder (persistent single WGP, 266 KB LDS, async double-buffered y_enc)
    k_decoder<<<1, 1024, 66560 * sizeof(float), stream>>>(ye, hsb, Wih_d, Whh_d,
                                                          bih_d, bhh_d, lin_w, lin_b, out);
}